// meta_model_36223754174517
// MI455X (gfx1250) — compile-verified
//
#include <hip/hip_runtime.h>
#include <math.h>

typedef float v2f __attribute__((ext_vector_type(2)));
typedef float v8f __attribute__((ext_vector_type(8)));

#define HH 256
#define WW 256
#define HW (HH*WW)
#define BB 4
#define LSTEPS 10
#define LEAKY 0.01f

#define KS 51
#define KR 25

#define TILE 16
#define HALOD 18      // TILE + 2 (conv2 needs 1-px halo of conv1 outputs)
#define NPAD 112      // 100 channels padded to 7*16
#define KPAD 20       // K=18 (2ch * 9 taps) padded to 5*4
#define MTOT 324      // 18*18 conv1 output pixels per tile
#define MPAD 336      // 21*16 (padded so all lanes store unconditionally)
#define MT 21         // M-tiles
#define ASTRIDE 17    // sAct row stride (coprime with 64 banks)

__device__ __forceinline__ int iclamp(int v, int lo, int hi) {
    return v < lo ? lo : (v > hi ? hi : v);
}

// ---------------------------------------------------------------------------
// init: images[0] = source, residuals[0] = broadcast(z0)
// ---------------------------------------------------------------------------
__global__ void init_kernel(const float* __restrict__ src,
                            const float* __restrict__ z0,
                            float* __restrict__ images0,
                            float* __restrict__ resid0) {
    int p = blockIdx.x * blockDim.x + threadIdx.x;
    if (p >= BB * HW) return;
    images0[p] = src[p];
    resid0[p]  = z0[p % HW];
}

// ---------------------------------------------------------------------------
// spatial gradient (edge pad) -> grads slot; tmp = -residual * g (2 channels)
// ---------------------------------------------------------------------------
__global__ void grad_mul_kernel(const float* __restrict__ img,
                                const float* __restrict__ res,
                                float* __restrict__ grads_out,  // (B,2,H,W)
                                float* __restrict__ tmp) {      // (B,2,H,W)
    int p = blockIdx.x * blockDim.x + threadIdx.x;
    if (p >= BB * HW) return;
    int b = p / HW;
    int yx = p - b * HW;
    int y = yx / WW, x = yx - y * WW;
    const float* I = img + (size_t)b * HW;
    int xm = iclamp(x - 1, 0, WW - 1), xp = iclamp(x + 1, 0, WW - 1);
    int ym = iclamp(y - 1, 0, HH - 1), yp = iclamp(y + 1, 0, HH - 1);
    float a00 = I[ym * WW + xm], a01 = I[ym * WW + x], a02 = I[ym * WW + xp];
    float a10 = I[y  * WW + xm],                        a12 = I[y  * WW + xp];
    float a20 = I[yp * WW + xm], a21 = I[yp * WW + x], a22 = I[yp * WW + xp];
    float gx = (a02 + 2.f * a12 + a22 - a00 - 2.f * a10 - a20) * 0.125f;
    float gy = (a20 + 2.f * a21 + a22 - a00 - 2.f * a01 - a02) * 0.125f;
    grads_out[((size_t)b * 2 + 0) * HW + yx] = gx;
    grads_out[((size_t)b * 2 + 1) * HW + yx] = gy;
    float r = -res[(size_t)b * HW + yx];
    tmp[((size_t)b * 2 + 0) * HW + yx] = r * gx;
    tmp[((size_t)b * 2 + 1) * HW + yx] = r * gy;
}

// ---------------------------------------------------------------------------
// separable 51-tap Gaussian blur (zero pad), sigma = 6
// ---------------------------------------------------------------------------
__global__ void blur_h_kernel(const float* __restrict__ in,
                              float* __restrict__ out) {
    __shared__ float sg[KS];
    __shared__ float sInv;
    int tid = threadIdx.x;
    if (tid < KS) { float d = (tid - KR) / 6.0f; sg[tid] = __expf(-0.5f * d * d); }
    __syncthreads();
    if (tid == 0) { float s = 0.f; for (int t = 0; t < KS; ++t) s += sg[t]; sInv = 1.0f / s; }
    __syncthreads();
    float inv = sInv;
    int p = blockIdx.x * blockDim.x + tid;
    if (p >= BB * 2 * HW) return;
    int img = p / HW;
    int yx = p - img * HW;
    int y = yx / WW, x = yx - y * WW;
    const float* row = in + (size_t)img * HW + (size_t)y * WW;
    float acc = 0.f;
    #pragma unroll 4
    for (int t = 0; t < KS; ++t) {
        int xx = x + t - KR;
        float v = (xx >= 0 && xx < WW) ? row[xx] : 0.f;
        acc += sg[t] * v;
    }
    out[p] = acc * inv;
}

__global__ void blur_v_kernel(const float* __restrict__ in,
                              float* __restrict__ field_out) { // NHWC (B,H,W,2)
    __shared__ float sg[KS];
    __shared__ float sInv;
    int tid = threadIdx.x;
    if (tid < KS) { float d = (tid - KR) / 6.0f; sg[tid] = __expf(-0.5f * d * d); }
    __syncthreads();
    if (tid == 0) { float s = 0.f; for (int t = 0; t < KS; ++t) s += sg[t]; sInv = 1.0f / s; }
    __syncthreads();
    float inv = sInv;
    int p = blockIdx.x * blockDim.x + tid;
    if (p >= BB * 2 * HW) return;
    int img = p / HW;
    int b = img >> 1, c = img & 1;
    int yx = p - img * HW;
    int y = yx / WW, x = yx - y * WW;
    const float* col = in + (size_t)img * HW;
    float acc = 0.f;
    #pragma unroll 4
    for (int t = 0; t < KS; ++t) {
        int yy = y + t - KR;
        float v = (yy >= 0 && yy < HH) ? col[yy * WW + x] : 0.f;
        acc += sg[t] * v;
    }
    field_out[(((size_t)b * HH + y) * WW + x) * 2 + c] = acc * inv;
}

// ---------------------------------------------------------------------------
// fused res-block: conv3x3(2->100)+bias -> leaky -> conv3x3(100->1), /10, +z
//
// One 16x16 output tile per workgroup (grid 16x16xB), 256 thr = 8 wave32.
// conv1 is an im2col GEMM (M=324 halo px -> 21 tiles, K=18->20, N=100->112)
// on v_wmma_f32_16x16x4_f32. The im2col A-panel and transposed weight panel
// are pre-staged zero-padded in LDS so the hot loop is branch-free:
// ds_load_2addr_b64 (A) + 5x v_wmma + pk bias/leaky + ds_store_2addr_b32.
// The 100-channel activation lives only in LDS; conv2 (100->1, N=1 so WMMA
// would waste 15/16 of the array) runs on VALU fp32 FMAs from LDS.
// ---------------------------------------------------------------------------
__global__ __launch_bounds__(256)
void resblock_kernel(const float* __restrict__ zcur,   // residuals[i] (B,HW)
                     const float* __restrict__ icur,   // images[i]    (B,HW)
                     const float* __restrict__ w1,     // (100,2,3,3)
                     const float* __restrict__ b1,     // (100)
                     const float* __restrict__ w2,     // (100,3,3)
                     float* __restrict__ znext) {      // residuals[i+1]
    __shared__ float sA[MPAD][KPAD];       // 26.9 KB zero-padded im2col panel
    __shared__ float sBt[NPAD][KPAD];      //  9.0 KB transposed conv1 weights
    __shared__ float sW2[100 * 9];         //  3.6 KB
    __shared__ float sBias[NPAD];          //  0.45 KB
    __shared__ float sAct[MPAD][ASTRIDE];  // 22.8 KB per-group activations

    int tid = threadIdx.x;
    int lane = tid & 31;
    // wave id as a provably wave-uniform (scalar) value: keeps the M-tile
    // loop on s_cmp/s_cbranch with EXEC untouched (WMMA wants EXEC all-1s).
    int wave = __builtin_amdgcn_readfirstlane(tid) >> 5;
    int b = blockIdx.z;
    int tx0 = blockIdx.x * TILE;
    int ty0 = blockIdx.y * TILE;

    const float* zb = zcur + (size_t)b * HW;
    const float* ib = icur + (size_t)b * HW;

    // ---- stage weights (zero-padded, transposed so K is contiguous) ----
    for (int idx = tid; idx < NPAD * KPAD; idx += 256) {
        int ch = idx / KPAD, k = idx - ch * KPAD;
        sBt[ch][k] = (k < 18 && ch < 100) ? w1[ch * 18 + k] : 0.f;
    }
    for (int idx = tid; idx < 900; idx += 256) sW2[idx] = w2[idx];
    for (int idx = tid; idx < NPAD; idx += 256) sBias[idx] = (idx < 100) ? b1[idx] : 0.f;

    // ---- build zero-padded im2col A panel straight from global ----
    for (int idx = tid; idx < MPAD * KPAD; idx += 256) {
        int m = idx / KPAD, k = idx - m * KPAD;
        float v = 0.f;
        if (m < MTOT && k < 18) {
            int ay = m / HALOD, ax = m - ay * HALOD;
            int ci = k / 9, tap = k - ci * 9;
            int dy = tap / 3, dx = tap - dy * 3;
            int gy = ty0 - 2 + ay + dy, gx = tx0 - 2 + ax + dx;
            if (gx >= 0 && gx < WW && gy >= 0 && gy < HH) {
                int o = gy * WW + gx;
                v = (ci == 0) ? zb[o] : ib[o];
            }
        }
        sA[m][k] = v;
    }
    __syncthreads();

    float c2acc = 0.f;
    int px = tid & 15, py = (tid >> 4) & 15;
    int n = lane & 15;
    int kh = (lane >> 4) << 1;   // lane half selects K pair within each 4-wide K step

    for (int grp = 0; grp < 7; ++grp) {
        // B fragments for this channel group: contiguous 8B-aligned LDS reads
        int ch = (grp << 4) + n;
        v2f bfrag[5];
        #pragma unroll
        for (int ks = 0; ks < 5; ++ks)
            bfrag[ks] = *(const v2f*)&sBt[ch][ks * 4 + kh];
        float bias = sBias[ch];

        for (int mt = wave; mt < MT; mt += 8) {
            int mrow = (mt << 4) + n;
            v2f afrag[5];
            #pragma unroll
            for (int ks = 0; ks < 5; ++ks)
                afrag[ks] = *(const v2f*)&sA[mrow][ks * 4 + kh];

            v8f acc = {};
            #pragma unroll
            for (int ks = 0; ks < 5; ++ks)
                acc = __builtin_amdgcn_wmma_f32_16x16x4_f32(
                    false, afrag[ks], false, bfrag[ks], (short)0, acc, false, false);

            // bias + leaky (max(x, 0.01x) -> v_pk_mul/v_pk_max) in D layout,
            // unconditional spill to LDS
            int mbase = (mt << 4) + ((lane < 16) ? 0 : 8);
            #pragma unroll
            for (int r = 0; r < 8; ++r) {
                float x = acc[r] + bias;
                sAct[mbase + r][n] = fmaxf(x, LEAKY * x);
            }
        }
        __syncthreads();
        // ---- conv2 partial sums over this channel group (VALU fp32) ----
        int cmax = (grp == 6) ? 4 : 16;   // channels 96..99 only in last group
        for (int c = 0; c < cmax; ++c) {
            const float* wr = &sW2[((grp << 4) + c) * 9];
            #pragma unroll
            for (int tap = 0; tap < 9; ++tap) {
                int dy = tap / 3, dx = tap - dy * 3;
                c2acc += wr[tap] * sAct[(py + dy) * HALOD + (px + dx)][c];
            }
        }
        __syncthreads();
    }
    int ox = tx0 + px, oy = ty0 + py;
    float zv = zb[oy * WW + ox];   // residuals[i] at this pixel
    znext[(size_t)b * HW + (size_t)oy * WW + ox] = zv + c2acc * 0.1f;
}

// ---------------------------------------------------------------------------
// bilinear warp by (id - field/10) + z_next * mu^2 / 10
// ---------------------------------------------------------------------------
__global__ void deform_kernel(const float* __restrict__ img_in,
                              const float* __restrict__ field,   // NHWC
                              const float* __restrict__ znext,
                              float* __restrict__ img_out) {
    int p = blockIdx.x * blockDim.x + threadIdx.x;
    if (p >= BB * HW) return;
    int b = p / HW;
    int yx = p - b * HW;
    int y = yx / WW, x = yx - y * WW;
    size_t fi = (((size_t)b * HH + y) * WW + x) * 2;
    float fx = (float)x - field[fi + 0] * 0.1f;
    float fy = (float)y - field[fi + 1] * 0.1f;
    float x0f = floorf(fx), y0f = floorf(fy);
    float wx = fx - x0f, wy = fy - y0f;
    int x0 = iclamp((int)x0f, 0, WW - 1);
    int x1 = iclamp((int)x0f + 1, 0, WW - 1);
    int y0 = iclamp((int)y0f, 0, HH - 1);
    int y1 = iclamp((int)y0f + 1, 0, HH - 1);
    const float* I = img_in + (size_t)b * HW;
    float Ia = I[y0 * WW + x0], Ib = I[y0 * WW + x1];
    float Ic = I[y1 * WW + x0], Id = I[y1 * WW + x1];
    float outv = (1.f - wx) * (1.f - wy) * Ia + wx * (1.f - wy) * Ib
               + (1.f - wx) * wy * Ic + wx * wy * Id;
    img_out[p] = outv + znext[p] * 1e-3f;   // MU^2 / L = 0.01/10
}

// ---------------------------------------------------------------------------
extern "C" void kernel_launch(void* const* d_in, const int* in_sizes, int n_in,
                              void* d_out, int out_size, void* d_ws, size_t ws_size,
                              hipStream_t stream) {
    const float* source = (const float*)d_in[0];
    const float* z0     = (const float*)d_in[1];
    const float* c1w    = (const float*)d_in[2]; // (10,100,2,3,3)
    const float* c1b    = (const float*)d_in[3]; // (10,100)
    const float* c2w    = (const float*)d_in[4]; // (10,1,100,3,3)
    float* out = (float*)d_out;

    const size_t nPix = (size_t)BB * HW;
    const size_t imagesOff = 0;
    const size_t fieldsOff = (size_t)(LSTEPS + 1) * nPix;
    const size_t residOff  = fieldsOff + (size_t)LSTEPS * nPix * 2;
    const size_t gradsOff  = residOff + (size_t)(LSTEPS + 1) * nPix;

    float* images = out + imagesOff;
    float* fields = out + fieldsOff;
    float* resid  = out + residOff;
    float* grads  = out + gradsOff;

    float* tmp  = (float*)d_ws;            // (B,2,H,W)
    float* tmpH = tmp + nPix * 2;          // (B,2,H,W)

    dim3 blk(256);
    dim3 gPix((unsigned)((nPix + 255) / 256));
    dim3 gPix2((unsigned)((2 * nPix + 255) / 256));
    dim3 gConv(WW / 16, HH / 16, BB);

    init_kernel<<<gPix, blk, 0, stream>>>(source, z0, images, resid);

    for (int i = 0; i < LSTEPS; ++i) {
        const float* imgI = images + (size_t)i * nPix;
        const float* resI = resid + (size_t)i * nPix;
        float* imgN = images + (size_t)(i + 1) * nPix;
        float* resN = resid + (size_t)(i + 1) * nPix;
        float* fieldI = fields + (size_t)i * nPix * 2;
        float* gradI  = grads + (size_t)i * nPix * 2;

        grad_mul_kernel<<<gPix, blk, 0, stream>>>(imgI, resI, gradI, tmp);
        blur_h_kernel<<<gPix2, blk, 0, stream>>>(tmp, tmpH);
        blur_v_kernel<<<gPix2, blk, 0, stream>>>(tmpH, fieldI);
        resblock_kernel<<<gConv, blk, 0, stream>>>(
            resI, imgI, c1w + (size_t)i * 1800, c1b + (size_t)i * 100,
            c2w + (size_t)i * 900, resN);
        deform_kernel<<<gPix, blk, 0, stream>>>(imgI, fieldI, resN, imgN);
    }
}